// ChannelPruner_29214367548114
// MI455X (gfx1250) — compile-verified
//
#include <hip/hip_runtime.h>

typedef float v2f __attribute__((ext_vector_type(2)));
typedef float v8f __attribute__((ext_vector_type(8)));

#define K_DIM    256
#define O_DIM    256
#define HW_DIM   3136   // 56*56
#define TILE_P   64
#define KH       128    // K half (double buffer)
#define PAIR_STR 160    // dwords per k-pair row: 128 data (64 px * 2) + 32 pad

// LDS layout (per half): element (k, p) -> dword (k>>1)*PAIR_STR + 2*p + (k&1)
// => a lane's WMMA B pair {X[k][p], X[k+1][p]} is one aligned 8B read,
//    and the two lane halves (k vs k+2 -> prow vs prow+1) are 32 banks apart.

#if __has_builtin(__builtin_amdgcn_global_load_async_to_lds_b32) && \
    __has_builtin(__builtin_amdgcn_s_wait_asynccnt)
#define USE_ASYNC_LDS 1
typedef __attribute__((address_space(1))) int as1_i32;   // global src
typedef __attribute__((address_space(3))) int as3_i32;   // LDS dst
#else
#define USE_ASYNC_LDS 0
#endif

__global__ __launch_bounds__(256)
void channel_pruner_gemm(const float* __restrict__ x,
                         const float* __restrict__ w,
                         float* __restrict__ out)
{
    // Two K-halves: [2][64 pair-rows][160 dwords] = 80 KB
    __shared__ float xlds[2][(KH / 2) * PAIR_STR];

    const int b    = blockIdx.y;
    const int p0   = blockIdx.x * TILE_P;
    const int tid  = threadIdx.x;
    const int lane = tid & 31;
    const int wave = tid >> 5;
    const int l15  = lane & 15;
    const int hi   = lane >> 4;          // 0: lanes 0-15, 1: lanes 16-31

    const float* xblk = x + (size_t)b * K_DIM * HW_DIM + p0;

#if USE_ASYNC_LDS
    // Async DMA staging with per-lane LDS scatter: each issue copies 128
    // contiguous bytes of one x-row (lanes = consecutive pixels) into the
    // k-pair-interleaved layout (32 distinct LDS banks). ASYNCcnt-tracked.
    auto stage_half = [&](int half, int kbase) {
        const int rbase = wave * 16;                          // 16 rows per wave
        #pragma unroll
        for (int i = 0; i < 16; ++i) {
            const int r = rbase + i;                          // row within half
            #pragma unroll
            for (int part = 0; part < 2; ++part) {
                const int p = part * 32 + lane;
                const float* g = xblk + (size_t)(kbase + r) * HW_DIM + p;
                float*       l = &xlds[half][(r >> 1) * PAIR_STR + 2 * p + (r & 1)];
                __builtin_amdgcn_global_load_async_to_lds_b32(
                    (as1_i32*)g, (as3_i32*)l, 0, 0);
            }
        }
    };
#else
    // Fallback: synchronous copy (global -> VGPR -> LDS), same layout.
    auto stage_half = [&](int half, int kbase) {
        const int r    = tid >> 1;                            // 128 rows, 256 thr
        const int part = (tid & 1) * 32;                      // half-row (32 f32)
        const float* src = xblk + (size_t)(kbase + r) * HW_DIM + part;
        float*       dst = &xlds[half][(r >> 1) * PAIR_STR + (r & 1)];
        #pragma unroll
        for (int q = 0; q < 32; q += 4) {
            float4 v = *(const float4*)(src + q);
            dst[2 * (part + q + 0)] = v.x;
            dst[2 * (part + q + 1)] = v.y;
            dst[2 * (part + q + 2)] = v.z;
            dst[2 * (part + q + 3)] = v.w;
        }
    };
#endif

    // Each wave: output channels [o0, o0+32) x 64 pixels = 2x4 WMMA tiles.
    const int o0 = wave * 32;

    // A-fragment (16x4 f32): lanes 0-15 row M=l15 hold {K=k0,k0+1},
    // lanes 16-31 hold {K=k0+2,k0+3} -> one b64 load per lane.
    const float* wRow0 = w + (size_t)(o0 + l15) * K_DIM + 2 * hi;
    const float* wRow1 = wRow0 + (size_t)16 * K_DIM;

    v8f acc[2][4];
    #pragma unroll
    for (int i = 0; i < 2; ++i)
        #pragma unroll
        for (int j = 0; j < 4; ++j)
            acc[i][j] = v8f{0.f, 0.f, 0.f, 0.f, 0.f, 0.f, 0.f, 0.f};

    auto compute_half = [&](int half, int kbase) {
        for (int kl = 0; kl < KH; kl += 4) {
            v2f a0 = *(const v2f*)(wRow0 + kbase + kl);
            v2f a1 = *(const v2f*)(wRow1 + kbase + kl);
            // B pair rows: lanes 0-15 use prow = kl/2, lanes 16-31 prow+1.
            const float* ldsrow = &xlds[half][((kl >> 1) + hi) * PAIR_STR + 2 * l15];
            #pragma unroll
            for (int j = 0; j < 4; ++j) {
                // one aligned 8B LDS read; j-tiles 128B apart -> ds_load_2addr_b64
                v2f bb = *(const v2f*)(ldsrow + 32 * j);
                acc[0][j] = __builtin_amdgcn_wmma_f32_16x16x4_f32(
                    false, a0, false, bb, (short)0, acc[0][j], false, false);
                acc[1][j] = __builtin_amdgcn_wmma_f32_16x16x4_f32(
                    false, a1, false, bb, (short)0, acc[1][j], false, false);
            }
        }
    };

#if USE_ASYNC_LDS
    stage_half(0, 0);
    __builtin_amdgcn_s_wait_asynccnt(0);   // own DMA done
    __syncthreads();                        // everyone's DMA done
    stage_half(1, KH);                      // fire-and-forget next half
    compute_half(0, 0);                     // 256 WMMAs hide the 32KB DMA
    __builtin_amdgcn_s_wait_asynccnt(0);
    __syncthreads();
    compute_half(1, KH);
#else
    stage_half(0, 0);
    stage_half(1, KH);
    __syncthreads();
    compute_half(0, 0);
    compute_half(1, KH);
#endif

    // D tiles: VGPR r holds row (r + 8*hi), col = l15 -> 64B row stores.
    float* outBase = out + (size_t)b * O_DIM * HW_DIM + p0;
    #pragma unroll
    for (int i = 0; i < 2; ++i) {
        const int otile = o0 + 16 * i;
        #pragma unroll
        for (int j = 0; j < 4; ++j) {
            #pragma unroll
            for (int r = 0; r < 8; ++r) {
                const int o = otile + r + 8 * hi;
                outBase[(size_t)o * HW_DIM + 16 * j + l15] = acc[i][j][r];
            }
        }
    }
}

extern "C" void kernel_launch(void* const* d_in, const int* in_sizes, int n_in,
                              void* d_out, int out_size, void* d_ws, size_t ws_size,
                              hipStream_t stream) {
    const float* x = (const float*)d_in[0];   // (32, 256, 56, 56) f32
    const float* w = (const float*)d_in[1];   // (256, 256, 1, 1)  f32
    float* out     = (float*)d_out;           // (32, 256, 56, 56) f32

    dim3 grid(HW_DIM / TILE_P, 32);           // 49 pixel tiles x 32 batches
    channel_pruner_gemm<<<grid, 256, 0, stream>>>(x, w, out);
}